// connect_loss_36215164240169
// MI455X (gfx1250) — compile-verified
//
#include <hip/hip_runtime.h>

// Problem constants (from setup_inputs): B=16, CH=8, H=W=512.
#define BB   16
#define CHN  8
#define HH   512
#define WW   512

// Tile: 64 cols x 16 rows per 256-thread block (4 pixels per thread).
#define TW   64
#define THR  16
// LDS tile with halo: 18 rows; cols c0-4 .. c0+67 -> 72-float row stride
// (left halo padded to 4 so every 16B chunk is aligned for async b128).
#define LROWS  18
#define LCOLS  72
#define LCH    (LROWS * LCOLS)            // 1296 floats per channel
#define LTOT   (CHN * LCH)                // 10368 floats = 41472 B
#define CHUNKS_PER_ROW (LCOLS / 4)        // 18 x b128 chunks
#define CHUNKS_PER_CH  (LROWS * CHUNKS_PER_ROW)   // 324
#define CHUNKS_TOT     (CHN * CHUNKS_PER_CH)      // 2592

__device__ __forceinline__ float sigm(float x) {
  return 1.0f / (1.0f + __expf(-x));
}
// -clip(log(v), -100): BCE term for 0/1 target after selecting v = t?p:(1-p)
__device__ __forceinline__ float nlogc(float v) {
  return -fmaxf(__logf(v), -100.0f);
}

__global__ __launch_bounds__(256)
void conn_loss_main(const float* __restrict__ cmap,
                    const int*   __restrict__ target,
                    const int*   __restrict__ contgt,
                    double*      __restrict__ scal,     // [4] bce, conn, bicon, dec
                    float*       __restrict__ dice_i,   // [B*W]
                    float*       __restrict__ dice_j,   // [B*W]
                    float*       __restrict__ dice_x)   // [B*W]
{
  __shared__ alignas(16) float sx[LTOT];  // raw logits, halo tile, 8 channels
  __shared__ float dsum[TW * 3];          // per-column dice partials (t, fp, t*fp)
  __shared__ float ssum[4];               // block scalar partials

  const int tx  = threadIdx.x;            // 0..15 -> 4 cols each
  const int ty  = threadIdx.y;            // 0..15 -> row
  const int tid = tx + ty * 16;
  const int b   = blockIdx.z;
  const int c0  = blockIdx.x * TW;
  const int r0  = blockIdx.y * THR;

  if (tid < TW * 3) dsum[tid] = 0.0f;
  if (tid < 4)      ssum[tid] = 0.0f;

  // ---- Stage 8-channel halo tile (raw logits) into LDS: async b128 loads ----
  // Chunk start coords are clamped into the plane (alignment preserved) so
  // clamped lanes never fault; logically-OOB neighbor values are forced to 0
  // at consumption, matching the zero-fill of the reference shift matrices.
  {
    const float* baseb   = cmap + (size_t)b * CHN * HH * WW;
    unsigned     sx_base = (unsigned)(size_t)(&sx[0]);  // LDS byte address
    for (int idx = tid; idx < CHUNKS_TOT; idx += 256) {
      int ch    = idx / CHUNKS_PER_CH;
      int rem   = idx - ch * CHUNKS_PER_CH;
      int row   = rem / CHUNKS_PER_ROW;
      int chunk = rem - row * CHUNKS_PER_ROW;
      int gr  = min(max(r0 - 1 + row, 0), HH - 1);
      int gcs = min(max(c0 - 4 + chunk * 4, 0), WW - 4);   // stays 4-aligned
      unsigned voff  = (unsigned)((((ch * HH) + gr) * WW + gcs) * 4);
      unsigned laddr = sx_base + (unsigned)(idx * 16);
      asm volatile("global_load_async_to_lds_b128 %0, %1, %2"
                   :: "v"(laddr), "v"(voff), "s"(baseb) : "memory");
    }
    asm volatile("s_wait_asynccnt 0x0" ::: "memory");
  }
  __syncthreads();

  // ---- Per-pixel compute: 4 consecutive columns per thread ----
  const int r     = r0 + ty;
  const int cbase = c0 + 4 * tx;

  const int4 tg4 = *reinterpret_cast<const int4*>(
      target + ((size_t)b * HH + r) * WW + cbase);
  const int t0[4] = {tg4.x, tg4.y, tg4.z, tg4.w};

  float maxv[4] = {-1.0f, -1.0f, -1.0f, -1.0f};
  float minv[4] = {2.0f, 2.0f, 2.0f, 2.0f};
  int   scon[4] = {0, 0, 0, 0};
  float conn_acc = 0.0f, bicon_acc = 0.0f;

  // neighbor offsets per vote channel ch (partner channel is 7-ch):
  // ch: 0:(-1,-1) 1:(-1,0) 2:(-1,+1) 3:(0,-1) 4:(0,+1) 5:(+1,-1) 6:(+1,0) 7:(+1,+1)
  constexpr int drt[8] = {-1,-1,-1, 0, 0, 1, 1, 1};
  constexpr int dct[8] = {-1, 0, 1,-1, 1,-1, 0, 1};

  #pragma unroll
  for (int ch = 0; ch < CHN; ++ch) {
    const int4 ct4 = *reinterpret_cast<const int4*>(
        contgt + (((size_t)b * CHN + ch) * HH + r) * WW + cbase);
    const int tc[4] = {ct4.x, ct4.y, ct4.z, ct4.w};

    const int qch = 7 - ch;
    const int dr  = drt[ch];
    const int dc  = dct[ch];

    // center logits: 16B-aligned float4 in LDS (col index 4 + 4*tx)
    const float4 xc4 = *reinterpret_cast<const float4*>(
        &sx[ch * LCH + (1 + ty) * LCOLS + (4 + 4 * tx)]);
    const float xc[4] = {xc4.x, xc4.y, xc4.z, xc4.w};

    const float* sq = &sx[qch * LCH + (1 + ty + dr) * LCOLS + (4 + 4 * tx + dc)];
    const bool rin = ((unsigned)(r + dr)) < (unsigned)HH;

    #pragma unroll
    for (int j = 0; j < 4; ++j) {
      float p = sigm(xc[j]);
      int   t = tc[j];
      conn_acc += nlogc(t ? p : (1.0f - p));

      bool in = rin && (((unsigned)(cbase + j + dc)) < (unsigned)WW);
      float q  = in ? sigm(sq[j]) : 0.0f;     // zero rolled in by shift matrices
      float bi = p * q;
      bicon_acc += nlogc(t ? bi : (1.0f - bi));

      maxv[j] = fmaxf(maxv[j], bi);
      minv[j] = fminf(minv[j], bi);
      scon[j] += t;
    }
  }

  float bce_acc = 0.0f, dec_acc = 0.0f;
  #pragma unroll
  for (int j = 0; j < 4; ++j) {
    float fp = maxv[j];
    bce_acc += nlogc(t0[j] ? fp : (1.0f - fp));

    bool  edge = (scon[j] > 0) && (scon[j] < 8);
    float pmin = edge ? minv[j] : 0.0f;
    dec_acc += nlogc(1.0f - pmin);

    int col = 4 * tx + j;
    atomicAdd(&dsum[col * 3 + 0], (float)t0[j]);
    atomicAdd(&dsum[col * 3 + 1], fp);
    atomicAdd(&dsum[col * 3 + 2], t0[j] ? fp : 0.0f);
  }

  // ---- wave32 reduction of the 4 scalar sums, then LDS, then f64 atomics ----
  float v[4] = {bce_acc, conn_acc, bicon_acc, dec_acc};
  #pragma unroll
  for (int off = 16; off > 0; off >>= 1) {
    #pragma unroll
    for (int k = 0; k < 4; ++k) v[k] += __shfl_down(v[k], off, 32);
  }
  if ((tid & 31) == 0) {
    #pragma unroll
    for (int k = 0; k < 4; ++k) atomicAdd(&ssum[k], v[k]);
  }
  __syncthreads();

  if (tid < TW) {
    int w = c0 + tid;
    atomicAdd(&dice_i[b * WW + w], dsum[tid * 3 + 0]);
    atomicAdd(&dice_j[b * WW + w], dsum[tid * 3 + 1]);
    atomicAdd(&dice_x[b * WW + w], dsum[tid * 3 + 2]);
  }
  if (tid == 0) {
    #pragma unroll
    for (int k = 0; k < 4; ++k) atomicAdd(&scal[k], (double)ssum[k]);
  }
}

__global__ __launch_bounds__(256)
void conn_loss_final(const double* __restrict__ scal,
                     const float*  __restrict__ dice_i,
                     const float*  __restrict__ dice_j,
                     const float*  __restrict__ dice_x,
                     float*        __restrict__ out)
{
  __shared__ float red[256];
  const int tid = threadIdx.x;
  float acc = 0.0f;
  for (int i = tid; i < BB * WW; i += 256) {
    float ii = dice_i[i], jj = dice_j[i], xx = dice_x[i];
    acc += 1.0f - (2.0f * xx + 0.001f) / (ii + jj + 0.001f);
  }
  red[tid] = acc;
  __syncthreads();
  for (int s = 128; s > 0; s >>= 1) {
    if (tid < s) red[tid] += red[tid + s];
    __syncthreads();
  }
  if (tid == 0) {
    const double n_small = (double)BB * HH * WW;           // 4194304
    const double n_big   = (double)BB * CHN * HH * WW;     // 33554432
    double dice  = (double)red[0] / (double)(BB * WW);
    double bce   = scal[0] / n_small;
    double conn  = scal[1] / n_big;
    double bicon = scal[2] / n_big;
    double dec   = scal[3] / n_small;
    out[0] = (float)(bce + conn + dice + 0.2 * bicon + dec);
  }
}

extern "C" void kernel_launch(void* const* d_in, const int* in_sizes, int n_in,
                              void* d_out, int out_size, void* d_ws, size_t ws_size,
                              hipStream_t stream) {
  const float* cmap   = (const float*)d_in[0];   // (16,8,512,512) f32
  const int*   target = (const int*)d_in[1];     // (16,1,512,512) i32
  const int*   contgt = (const int*)d_in[2];     // (16,8,512,512) i32
  // d_in[3], d_in[4] (hori/verti shift matrices) are implemented as index shifts.

  double* scal   = (double*)d_ws;                         // 4 doubles
  float*  dice_i = (float*)((char*)d_ws + 32);            // 8192 floats
  float*  dice_j = dice_i + BB * WW;
  float*  dice_x = dice_j + BB * WW;
  const size_t ws_used = 32 + (size_t)3 * BB * WW * sizeof(float);

  hipMemsetAsync(d_ws, 0, ws_used, stream);

  dim3 grid(WW / TW, HH / THR, BB);   // (8, 32, 16)
  dim3 block(16, 16);
  conn_loss_main<<<grid, block, 0, stream>>>(cmap, target, contgt,
                                             scal, dice_i, dice_j, dice_x);
  conn_loss_final<<<1, 256, 0, stream>>>(scal, dice_i, dice_j, dice_x,
                                         (float*)d_out);
}